// TriPlane_29368986370659
// MI455X (gfx1250) — compile-verified
//
#include <hip/hip_runtime.h>
#include <cstdint>

// ---------------- problem constants ----------------
#define NPTS      1000000
#define CDIM      32
#define HW_C      (100*100)     // coarse plane positions
#define HW_F      (400*400)     // fine plane positions
#define W_C       100
#define W_F       400

// scratch layout (floats): 3 coarse planes [HW_C*32] then 3 fine planes [HW_F*32]
#define WS_COARSE_STRIDE (HW_C * CDIM)
#define WS_FINE_STRIDE   (HW_F * CDIM)
#define WS_FINE_BASE     (3 * WS_COARSE_STRIDE)
#define WS_FLOATS        (3 * WS_COARSE_STRIDE + 3 * WS_FINE_STRIDE)

// =====================================================================
// Stage kernel: transpose one plane [C=32, HW] -> [HW, C=32] in scratch.
// Phase 1 uses gfx1250 async global->LDS loads (ASYNCcnt), phase 2 writes
// transposed, fully coalesced 128B rows.
// =====================================================================
__global__ __launch_bounds__(256) void stage_plane_kernel(
    const float* __restrict__ src,   // [32, HW]
    float*       __restrict__ dst,   // [HW, 32]
    int HW)
{
    __shared__ float tile[64][33];   // pad to 33 words: conflict-free both phases

    const int t   = threadIdx.x;
    const int p0  = blockIdx.x * 64;

    // ---- phase 1: async global -> LDS, coalesced reads of src rows ----
    {
        const int pos = t & 63;          // lanes of a wave sweep 32 consecutive pos
        const int c0  = (t >> 6) * 8;    // 4 thread-groups x 8 channels each
        const int gp  = p0 + pos;
        if (gp < HW) {
#pragma unroll
            for (int j = 0; j < 8; ++j) {
                const int c = c0 + j;
                unsigned lds_addr = (unsigned)(uintptr_t)(&tile[pos][c]);
                uint64_t gaddr    = (uint64_t)(uintptr_t)(src + (size_t)c * HW + gp);
                // GLOBAL_LOAD_ASYNC_TO_LDS_B32: VDST = LDS byte addr, VADDR = 64b global addr
                asm volatile("global_load_async_to_lds_b32 %0, %1, off"
                             :
                             : "v"(lds_addr), "v"(gaddr)
                             : "memory");
            }
        }
    }
    __builtin_amdgcn_s_wait_asynccnt(0);
    __syncthreads();

    // ---- phase 2: coalesced [pos, C] stores (wave = 32 channels of one pos) ----
    {
        const int c  = t & 31;
        const int pw = t >> 5;           // 0..7
#pragma unroll
        for (int j = 0; j < 8; ++j) {
            const int pp = pw * 8 + j;   // 0..63
            const int gp = p0 + pp;
            if (gp < HW)
                dst[(size_t)gp * CDIM + c] = tile[pp][c];
        }
    }
}

// =====================================================================
// Bilinear sample from a [HW, 32] staged plane; lane = channel.
// W is compile-time so row math folds to immediates.
// =====================================================================
template <int W>
__device__ __forceinline__ float sample_staged(
    const float* __restrict__ base, float a, float b, int lane)
{
    constexpr int   Wm1   = W - 1;
    constexpr float wm1f  = (float)Wm1;
    constexpr float scale = 0.125f * wm1f;   // u = (a*0.25+1)*0.5*(W-1)
    constexpr float off   = 0.5f  * wm1f;

    float u = fmaf(a, scale, off);
    float v = fmaf(b, scale, off);
    u = fminf(fmaxf(u, 0.0f), wm1f);
    v = fminf(fmaxf(v, 0.0f), wm1f);
    const float uf = floorf(u), vf = floorf(v);
    const int x0 = (int)uf, y0 = (int)vf;
    const int x1 = min(x0 + 1, Wm1);
    const int y1 = min(y0 + 1, Wm1);
    const float wx = u - uf, wy = v - vf;
    const int r0 = y0 * W, r1 = y1 * W;

    // each corner: 32 lanes read 32 consecutive floats = one full cacheline
    const float v00 = base[(r0 + x0) * CDIM + lane];
    const float v01 = base[(r0 + x1) * CDIM + lane];
    const float v10 = base[(r1 + x0) * CDIM + lane];
    const float v11 = base[(r1 + x1) * CDIM + lane];

    const float top = fmaf(wx, v01 - v00, v00);
    const float bot = fmaf(wx, v11 - v10, v10);
    return fmaf(wy, bot - top, top);
}

__device__ __forceinline__ void point_features(
    const float* __restrict__ ws, float x, float y, float z, int lane,
    float& coarse, float& fine)
{
    const float* cxy = ws;
    const float* cxz = ws + 1 * WS_COARSE_STRIDE;
    const float* cyz = ws + 2 * WS_COARSE_STRIDE;
    const float* fxy = ws + WS_FINE_BASE;
    const float* fxz = ws + WS_FINE_BASE + 1 * WS_FINE_STRIDE;
    const float* fyz = ws + WS_FINE_BASE + 2 * WS_FINE_STRIDE;
    // sample args: (a indexes W, b indexes H) per reference sample_plane(plane, gx, gy)
    coarse = sample_staged<W_C>(cxy, x, y, lane)
           + sample_staged<W_C>(cxz, x, z, lane)
           + sample_staged<W_C>(cyz, y, z, lane);
    fine   = sample_staged<W_F>(fxy, x, y, lane)
           + sample_staged<W_F>(fxz, x, z, lane)
           + sample_staged<W_F>(fyz, y, z, lane);
}

// =====================================================================
// Fast path: one wave per point pair (2 points/wave), lane = channel.
// Branch-free main path so all 48 gathers of both points pipeline.
// NT loads for xyz (streamed once), NT stores for out (256 MB stream must
// not evict the L2-resident planes).
// =====================================================================
__global__ __launch_bounds__(256) void triplane_fast_kernel(
    const float* __restrict__ xyz,
    const float* __restrict__ ws,
    float*       __restrict__ out,
    int npts)
{
    const int wave = blockIdx.x * (blockDim.x >> 5) + (threadIdx.x >> 5);
    const int lane = threadIdx.x & 31;
    const int p0   = wave * 2;

    // warm upcoming xyz (emits global_prefetch_b8)
    if (p0 + 512 < npts) __builtin_prefetch(xyz + (size_t)3 * (p0 + 512), 0, 1);

    if (p0 + 1 < npts) {
        // common path: both points, no intervening control flow
        const float x0p = __builtin_nontemporal_load(xyz + 3 * (size_t)p0 + 0);
        const float y0p = __builtin_nontemporal_load(xyz + 3 * (size_t)p0 + 1);
        const float z0p = __builtin_nontemporal_load(xyz + 3 * (size_t)p0 + 2);
        const float x1p = __builtin_nontemporal_load(xyz + 3 * (size_t)p0 + 3);
        const float y1p = __builtin_nontemporal_load(xyz + 3 * (size_t)p0 + 4);
        const float z1p = __builtin_nontemporal_load(xyz + 3 * (size_t)p0 + 5);

        float c0, f0, c1, f1;
        point_features(ws, x0p, y0p, z0p, lane, c0, f0);
        point_features(ws, x1p, y1p, z1p, lane, c1, f1);

        float* o = out + (size_t)p0 * 64;
        __builtin_nontemporal_store(c0, o + lane);
        __builtin_nontemporal_store(f0, o + 32 + lane);
        __builtin_nontemporal_store(c1, o + 64 + lane);
        __builtin_nontemporal_store(f1, o + 96 + lane);
    } else if (p0 < npts) {
        const float x = __builtin_nontemporal_load(xyz + 3 * (size_t)p0 + 0);
        const float y = __builtin_nontemporal_load(xyz + 3 * (size_t)p0 + 1);
        const float z = __builtin_nontemporal_load(xyz + 3 * (size_t)p0 + 2);
        float c, f;
        point_features(ws, x, y, z, lane, c, f);
        __builtin_nontemporal_store(c, out + (size_t)p0 * 64 + lane);
        __builtin_nontemporal_store(f, out + (size_t)p0 * 64 + 32 + lane);
    }
}

// =====================================================================
// Fallback (ws too small): gather straight from [C,H,W] planes.
// Correct but uncoalesced (lane = channel, stride H*W across lanes).
// =====================================================================
template <int W, int HW>
__device__ __forceinline__ float sample_raw(
    const float* __restrict__ plane, float a, float b, int lane)
{
    constexpr int   Wm1   = W - 1;
    constexpr float wm1f  = (float)Wm1;
    constexpr float scale = 0.125f * wm1f;
    constexpr float off   = 0.5f  * wm1f;

    float u = fmaf(a, scale, off);
    float v = fmaf(b, scale, off);
    u = fminf(fmaxf(u, 0.0f), wm1f);
    v = fminf(fmaxf(v, 0.0f), wm1f);
    const float uf = floorf(u), vf = floorf(v);
    const int x0 = (int)uf, y0 = (int)vf;
    const int x1 = min(x0 + 1, Wm1);
    const int y1 = min(y0 + 1, Wm1);
    const float wx = u - uf, wy = v - vf;
    const float* ch = plane + (size_t)lane * HW;
    const float v00 = ch[y0 * W + x0];
    const float v01 = ch[y0 * W + x1];
    const float v10 = ch[y1 * W + x0];
    const float v11 = ch[y1 * W + x1];
    const float top = fmaf(wx, v01 - v00, v00);
    const float bot = fmaf(wx, v11 - v10, v10);
    return fmaf(wy, bot - top, top);
}

__global__ __launch_bounds__(256) void triplane_direct_kernel(
    const float* __restrict__ xyz,
    const float* __restrict__ cxy, const float* __restrict__ cxz, const float* __restrict__ cyz,
    const float* __restrict__ fxy, const float* __restrict__ fxz, const float* __restrict__ fyz,
    float* __restrict__ out, int npts)
{
    const int wave = blockIdx.x * (blockDim.x >> 5) + (threadIdx.x >> 5);
    const int lane = threadIdx.x & 31;
    const int p    = wave;            // 1 point per wave in fallback
    if (p >= npts) return;

    const float x = xyz[3 * (size_t)p + 0];
    const float y = xyz[3 * (size_t)p + 1];
    const float z = xyz[3 * (size_t)p + 2];
    const float coarse =
        sample_raw<W_C, HW_C>(cxy, x, y, lane) +
        sample_raw<W_C, HW_C>(cxz, x, z, lane) +
        sample_raw<W_C, HW_C>(cyz, y, z, lane);
    const float fine =
        sample_raw<W_F, HW_F>(fxy, x, y, lane) +
        sample_raw<W_F, HW_F>(fxz, x, z, lane) +
        sample_raw<W_F, HW_F>(fyz, y, z, lane);
    __builtin_nontemporal_store(coarse, out + (size_t)p * 64 + lane);
    __builtin_nontemporal_store(fine,   out + (size_t)p * 64 + 32 + lane);
}

// =====================================================================
extern "C" void kernel_launch(void* const* d_in, const int* in_sizes, int n_in,
                              void* d_out, int out_size, void* d_ws, size_t ws_size,
                              hipStream_t stream)
{
    const float* xyz = (const float*)d_in[0];
    const float* pc[3] = { (const float*)d_in[1], (const float*)d_in[2], (const float*)d_in[3] };
    const float* pf[3] = { (const float*)d_in[4], (const float*)d_in[5], (const float*)d_in[6] };
    float* out = (float*)d_out;

    const int npts = in_sizes[0] / 3;
    const size_t ws_needed = (size_t)WS_FLOATS * sizeof(float);

    if (ws_size >= ws_needed) {
        float* ws = (float*)d_ws;
        // stage (transpose) all six planes into [HW, C] layout
        for (int i = 0; i < 3; ++i) {
            stage_plane_kernel<<<(HW_C + 63) / 64, 256, 0, stream>>>(
                pc[i], ws + (size_t)i * WS_COARSE_STRIDE, HW_C);
        }
        for (int i = 0; i < 3; ++i) {
            stage_plane_kernel<<<(HW_F + 63) / 64, 256, 0, stream>>>(
                pf[i], ws + WS_FINE_BASE + (size_t)i * WS_FINE_STRIDE, HW_F);
        }
        const int waves  = (npts + 1) / 2;          // 2 points per wave
        const int blocks = (waves + 7) / 8;         // 8 waves per block
        triplane_fast_kernel<<<blocks, 256, 0, stream>>>(xyz, ws, out, npts);
    } else {
        const int blocks = (npts + 7) / 8;          // 1 point per wave fallback
        triplane_direct_kernel<<<blocks, 256, 0, stream>>>(
            xyz, pc[0], pc[1], pc[2], pf[0], pf[1], pf[2], out, npts);
    }
}